// FusedQwenLowBitMultiDecoderlayer_28355374088642
// MI455X (gfx1250) — compile-verified
//
#include <hip/hip_runtime.h>
#include <hip/hip_bf16.h>
#include <math.h>

typedef __attribute__((ext_vector_type(2))) float v2f;
typedef __attribute__((ext_vector_type(8))) float v8f;

#define HID   2048
#define INTER 5632
#define NHEAD 16
#define NKV   2
#define HD    128
#define BATCH 4
#define TSEQ  1025

#define CK    128            // K-chunk per double-buffer stage (128 rows x 128 cols x 4B = 64KB)
#define WSTR  132            // LDS row stride for weight chunk (132 % 64 == 4 -> conflict-free)
#define MAXSLICE 512         // max K-slice per block (act panel staged in LDS)

// ---- CDNA5 async copy to LDS + counter wait (inline asm: toolchain-portable) ----
#if defined(__has_builtin) && __has_builtin(__builtin_amdgcn_s_wait_asynccnt)
#define WAIT_ASYNC(n) __builtin_amdgcn_s_wait_asynccnt(n)
#else
#define WAIT_ASYNC(n) asm volatile("s_wait_asynccnt %0" ::"i"(n) : "memory")
#endif

#define ASYNC_LOAD_B128(ldsoff, gaddr)                                              \
  asm volatile("global_load_async_to_lds_b128 %0, %1, off" ::"v"(ldsoff),           \
               "v"(gaddr)                                                           \
               : "memory")

// ---------------------------------------------------------------------------
// WMMA f32 16x16x4:  D[M=16 outs, N=16 batch-cols] += A(W tile) x B(acts)
// A (16x4): lane L<16 holds row M=L, K={0,1}; lane L>=16 holds K={2,3}
// B (4x16): lane L holds col N=L%16 (batch; cols>=4 are don't-care), K={half*2,+1}
// ---------------------------------------------------------------------------
__device__ __forceinline__ v8f wmma_kloop(v8f acc, const float* wrow, const float* arow,
                                          int len) {
#pragma unroll 8
  for (int k = 0; k < len; k += 4) {
    v2f a = *(const v2f*)(wrow + k);
    v2f b = *(const v2f*)(arow + k);
    acc = __builtin_amdgcn_wmma_f32_16x16x4_f32(false, a, false, b, (short)0, acc,
                                                false, false);
  }
  return acc;
}

// ---------------------------------------------------------------------------
// Generic split-K batched-GEMV with async-to-LDS weight streaming.
// Block = 256 threads = 8 waves; block owns 128 output rows x one K-slice.
// Up to 3 concatenated weight segments (rows [0,n0) -> W0, [n0,n0+n1) -> W1, rest W2).
// Writes fp32 partials: pbuf[(ks*Ntot + m)*4 + b].
// ---------------------------------------------------------------------------
__global__ __launch_bounds__(256) void gemv_async_kernel(
    const float* __restrict__ W0, const float* __restrict__ W1,
    const float* __restrict__ W2, int n0, int n1, const float* __restrict__ act, int K,
    int slice, float* __restrict__ pbuf, int Ntot) {
  __shared__ __align__(16) float wbuf[2][128 * WSTR];
  __shared__ __align__(16) float sact[BATCH * (MAXSLICE + 4)];

  const int tid = threadIdx.x, wv = tid >> 5, ln = tid & 31;
  const int half = ln >> 4, l16 = ln & 15, bcol = ln & 3;
  const int m0 = blockIdx.x * 128;
  const int ks = blockIdx.y;
  const int klo = ks * slice;
  const int ASTR = slice + 4;  // slice % 64 == 0 -> stride % 64 == 4: conflict-free

  const float* W; int rowbase;
  if (m0 < n0)           { W = W0; rowbase = m0; }
  else if (m0 < n0 + n1) { W = W1; rowbase = m0 - n0; }
  else                   { W = W2; rowbase = m0 - n0 - n1; }
  const float* gbase = W + (size_t)rowbase * K + klo;

  const unsigned ldsb0 = (unsigned)(unsigned long long)(const void*)&wbuf[0][0];
  const unsigned ldsb1 = (unsigned)(unsigned long long)(const void*)&wbuf[1][0];

  // Issue one 64KB weight chunk: wave w DMAs rows {w, w+8, ..., w+120}; each
  // instruction moves 32 lanes x 16B = one fully-coalesced 512B row.
#define ISSUE_CHUNK(cidx, bsel)                                                     \
  do {                                                                              \
    const float* _g = gbase + (cidx)*CK + ln * 4;                                   \
    const unsigned _lb = ((bsel) ? ldsb1 : ldsb0) + (unsigned)(ln * 16);            \
    _Pragma("unroll") for (int _it = 0; _it < 16; ++_it) {                          \
      const int _row = _it * 8 + wv;                                                \
      const unsigned long long _ga =                                                \
          (unsigned long long)(const void*)(_g + (size_t)_row * K);                 \
      const unsigned _lo = _lb + (unsigned)(_row * (WSTR * 4));                     \
      ASYNC_LOAD_B128(_lo, _ga);                                                    \
    }                                                                               \
  } while (0)

  ISSUE_CHUNK(0, 0);  // prime the pipeline

  // Stage activation panel [4][slice] for this K-slice (tiny: <=8KB).
  for (int b4 = 0; b4 < BATCH; ++b4)
    for (int c = tid; c < slice; c += 256)
      sact[b4 * ASTR + c] = act[(size_t)b4 * K + klo + c];

  v8f acc = {0.f, 0.f, 0.f, 0.f, 0.f, 0.f, 0.f, 0.f};
  const int nch = slice / CK;
  for (int c = 0; c < nch; ++c) {
    if (c + 1 < nch) {
      ISSUE_CHUNK(c + 1, (c + 1) & 1);
      WAIT_ASYNC(16);  // chunk c landed (16 newer loads may remain in flight)
    } else {
      WAIT_ASYNC(0);
    }
    __syncthreads();  // whole buffer (all waves' DMAs + act panel) visible
    const float* wfrag = &wbuf[c & 1][(wv * 16 + l16) * WSTR + half * 2];
    const float* afrag = sact + bcol * ASTR + c * CK + half * 2;
    acc = wmma_kloop(acc, wfrag, afrag, CK);
    __syncthreads();  // all waves done reading before this buffer is re-targeted
  }
#undef ISSUE_CHUNK

  if (l16 < BATCH) {
    const int mb = m0 + wv * 16;
#pragma unroll
    for (int r = 0; r < 8; ++r) {
      const int m = mb + r + half * 8;  // D layout: VGPR r = M=r (lanes<16) / r+8
      pbuf[((size_t)ks * Ntot + m) * 4 + l16] = acc[r];
    }
  }
}

// ---- split-K reduction epilogues (deterministic: fixed summation order) ----
__global__ __launch_bounds__(256) void qkv_reduce_kernel(
    const float* __restrict__ pbuf, const float* __restrict__ qb,
    const float* __restrict__ kb, const float* __restrict__ vb, float* __restrict__ qo,
    float* __restrict__ ko, float* __restrict__ vo) {
  const int idx = blockIdx.x * 256 + threadIdx.x;  // [0, 2560*4)
  if (idx >= 2560 * BATCH) return;
  const int m = idx >> 2, b = idx & 3;
  float s = 0.f;
#pragma unroll
  for (int ks = 0; ks < 4; ++ks) s += pbuf[((size_t)ks * 2560 + m) * 4 + b];
  if (m < HID)            qo[b * HID + m] = s + qb[m];
  else if (m < HID + 256) ko[b * 256 + (m - HID)] = s + kb[m - HID];
  else                    vo[b * 256 + (m - HID - 256)] = s + vb[m - HID - 256];
}

__global__ __launch_bounds__(256) void resid_reduce_kernel(const float* __restrict__ pbuf,
                                                           int KS, float* __restrict__ h) {
  const int idx = blockIdx.x * 256 + threadIdx.x;  // [0, 2048*4)
  const int m = idx >> 2, b = idx & 3;
  float s = 0.f;
  for (int ks = 0; ks < KS; ++ks) s += pbuf[((size_t)ks * HID + m) * 4 + b];
  h[b * HID + m] += s;
}

__global__ __launch_bounds__(256) void gu_reduce_kernel(const float* __restrict__ pbuf,
                                                        float* __restrict__ actm) {
  const int idx = blockIdx.x * 256 + threadIdx.x;  // [0, 5632*4)
  const int j = idx >> 2, b = idx & 3;
  float g = 0.f, u = 0.f;
#pragma unroll
  for (int ks = 0; ks < 4; ++ks) {
    g += pbuf[((size_t)ks * (2 * INTER) + j) * 4 + b];
    u += pbuf[((size_t)ks * (2 * INTER) + INTER + j) * 4 + b];
  }
  actm[b * INTER + j] = (g / (1.f + __expf(-g))) * u;  // silu(g)*u
}

// ---------------------------------------------------------------------------
__global__ __launch_bounds__(256) void copy_kernel(const float* __restrict__ src,
                                                   float* __restrict__ dst) {
  const int i = blockIdx.x * 256 + threadIdx.x;
  if (i < BATCH * HID) dst[i] = src[i];
}

__global__ __launch_bounds__(256) void rmsnorm_kernel(const float* __restrict__ h,
                                                      const float* __restrict__ w,
                                                      float* __restrict__ xn) {
  const int b = blockIdx.x, tid = threadIdx.x;
  __shared__ float red[8];
  float s = 0.f;
  for (int c = tid; c < HID; c += 256) { float v = h[b * HID + c]; s += v * v; }
  for (int off = 16; off > 0; off >>= 1) s += __shfl_xor(s, off, 32);
  if ((tid & 31) == 0) red[tid >> 5] = s;
  __syncthreads();
  float tot = 0.f;
#pragma unroll
  for (int j = 0; j < 8; ++j) tot += red[j];
  const float rs = rsqrtf(tot / (float)HID + 1e-6f);
  for (int c = tid; c < HID; c += 256) xn[b * HID + c] = h[b * HID + c] * rs * w[c];
}

__global__ __launch_bounds__(64) void rope_kernel(float* __restrict__ q,
                                                  float* __restrict__ k,
                                                  const float* __restrict__ ccos,
                                                  const float* __restrict__ csin,
                                                  const long long* __restrict__ pos) {
  const int b = blockIdx.x / (NHEAD + NKV);
  const int r = blockIdx.x % (NHEAD + NKV);
  const int p = (int)pos[b];
  const float* co = ccos + (size_t)p * HD;
  const float* si = csin + (size_t)p * HD;
  float* base = (r < NHEAD) ? (q + b * HID + r * HD)
                            : (k + b * (NKV * HD) + (r - NHEAD) * HD);
  const int d = threadIdx.x;  // 0..63
  const float x1 = base[d], x2 = base[d + 64];
  base[d] = x1 * co[d] - x2 * si[d];
  base[d + 64] = x2 * co[d + 64] + x1 * si[d + 64];
}

__global__ __launch_bounds__(256) void attn_kernel(
    const float* __restrict__ q, const float* __restrict__ knew,
    const float* __restrict__ vnew, const float* __restrict__ pastK,
    const float* __restrict__ pastV, const float* __restrict__ mask,
    float* __restrict__ aout) {
  const int b = blockIdx.x >> 4, h = blockIdx.x & 15;
  const int kvh = h >> 3;
  __shared__ float sc[TSEQ];
  __shared__ float red[8];
  __shared__ float vpart[8][HD];
  const int tid = threadIdx.x, wave = tid >> 5, lane = tid & 31;
  const float scale = 0.08838834764831845f;  // 1/sqrt(128)

  const float4 qv = *(const float4*)(q + b * HID + h * HD + lane * 4);
  const float* Kb = pastK + ((size_t)(b * NKV + kvh)) * 1024 * HD;
  const float* kn = knew + b * (NKV * HD) + kvh * HD;

  for (int t = wave; t < TSEQ; t += 8) {
    const float* kr = (t < 1024) ? (Kb + (size_t)t * HD) : kn;
    const float4 k4 = *(const float4*)(kr + lane * 4);
    float d = qv.x * k4.x + qv.y * k4.y + qv.z * k4.z + qv.w * k4.w;
    for (int off = 16; off > 0; off >>= 1) d += __shfl_xor(d, off, 32);
    if (lane == 0) sc[t] = d * scale + mask[b * TSEQ + t];
  }
  __syncthreads();

  float m = -1e30f;
  for (int t = tid; t < TSEQ; t += 256) m = fmaxf(m, sc[t]);
  for (int off = 16; off > 0; off >>= 1) m = fmaxf(m, __shfl_xor(m, off, 32));
  if (lane == 0) red[wave] = m;
  __syncthreads();
  float mm = red[0];
#pragma unroll
  for (int j = 1; j < 8; ++j) mm = fmaxf(mm, red[j]);
  __syncthreads();

  float s = 0.f;
  for (int t = tid; t < TSEQ; t += 256) { float e = __expf(sc[t] - mm); sc[t] = e; s += e; }
  for (int off = 16; off > 0; off >>= 1) s += __shfl_xor(s, off, 32);
  if (lane == 0) red[wave] = s;
  __syncthreads();
  float tot = 0.f;
#pragma unroll
  for (int j = 0; j < 8; ++j) tot += red[j];
  const float inv = 1.f / tot;

  const float* Vb = pastV + ((size_t)(b * NKV + kvh)) * 1024 * HD;
  const float* vn = vnew + b * (NKV * HD) + kvh * HD;
  float4 acc = {0.f, 0.f, 0.f, 0.f};
  for (int t = wave; t < TSEQ; t += 8) {
    const float* vr = (t < 1024) ? (Vb + (size_t)t * HD) : vn;
    const float4 v4 = *(const float4*)(vr + lane * 4);
    const float p = sc[t];
    acc.x += p * v4.x; acc.y += p * v4.y; acc.z += p * v4.z; acc.w += p * v4.w;
  }
  vpart[wave][lane * 4 + 0] = acc.x; vpart[wave][lane * 4 + 1] = acc.y;
  vpart[wave][lane * 4 + 2] = acc.z; vpart[wave][lane * 4 + 3] = acc.w;
  __syncthreads();
  if (tid < HD) {
    float o = 0.f;
#pragma unroll
    for (int j = 0; j < 8; ++j) o += vpart[j][tid];
    aout[b * HID + h * HD + tid] = o * inv;
  }
}

// ---------------------------------------------------------------------------
extern "C" void kernel_launch(void* const* d_in, const int* in_sizes, int n_in,
                              void* d_out, int out_size, void* d_ws, size_t ws_size,
                              hipStream_t stream) {
  (void)in_sizes; (void)n_in; (void)out_size; (void)ws_size;
  const float* x     = (const float*)d_in[0];
  const float* mask  = (const float*)d_in[1];
  const float* ccos  = (const float*)d_in[2];
  const float* csin  = (const float*)d_in[3];
  const float* ln1   = (const float*)d_in[4];
  const float* ln2   = (const float*)d_in[5];
  const float* Wq    = (const float*)d_in[6];
  const float* qbias = (const float*)d_in[7];
  const float* Wk    = (const float*)d_in[8];
  const float* kbias = (const float*)d_in[9];
  const float* Wv    = (const float*)d_in[10];
  const float* vbias = (const float*)d_in[11];
  const float* Wo    = (const float*)d_in[12];
  const float* Wg    = (const float*)d_in[13];
  const float* Wu    = (const float*)d_in[14];
  const float* Wd    = (const float*)d_in[15];
  const float* pk    = (const float*)d_in[16];
  const float* pv    = (const float*)d_in[17];
  const long long* pos = (const long long*)d_in[18];

  float* h  = (float*)d_out;  // running hidden state [4][2048]
  float* ws = (float*)d_ws;
  float* xn   = ws;            // 8192
  float* qb   = ws + 8192;     // 8192
  float* kb   = ws + 16384;    // 1024
  float* vb   = ws + 17408;    // 1024
  float* ao   = ws + 18432;    // 8192
  float* actm = ws + 26624;    // 22528
  float* pbuf = ws + 49152;    // up to 180224 (GU partials)

  const size_t sWqo = (size_t)HID * HID;
  const size_t sWkv = (size_t)(NKV * HD) * HID;
  const size_t sWgu = (size_t)INTER * HID;
  const size_t sWd  = (size_t)HID * INTER;
  const size_t sKV  = (size_t)BATCH * NKV * 1024 * HD;

  copy_kernel<<<32, 256, 0, stream>>>(x, h);
  for (int i = 0; i < 4; ++i) {
    rmsnorm_kernel<<<4, 256, 0, stream>>>(h, ln1 + (size_t)i * HID, xn);
    // fused QKV: Ntot=2560, K=2048, split-K x4 (slice 512) -> 80 blocks
    gemv_async_kernel<<<dim3(20, 4), 256, 0, stream>>>(
        Wq + i * sWqo, Wk + i * sWkv, Wv + i * sWkv, HID, 256, xn, HID, 512, pbuf, 2560);
    qkv_reduce_kernel<<<40, 256, 0, stream>>>(pbuf, qbias + (size_t)i * HID,
                                              kbias + (size_t)i * NKV * HD,
                                              vbias + (size_t)i * NKV * HD, qb, kb, vb);
    rope_kernel<<<BATCH * (NHEAD + NKV), 64, 0, stream>>>(qb, kb, ccos, csin, pos);
    attn_kernel<<<BATCH * NHEAD, 256, 0, stream>>>(qb, kb, vb, pk + i * sKV,
                                                   pv + i * sKV, mask, ao);
    // O-proj: Ntot=2048, K=2048, split-K x8 (slice 256) -> 128 blocks
    gemv_async_kernel<<<dim3(16, 8), 256, 0, stream>>>(
        Wo + i * sWqo, Wo + i * sWqo, Wo + i * sWqo, HID, 0, ao, HID, 256, pbuf, HID);
    resid_reduce_kernel<<<32, 256, 0, stream>>>(pbuf, 8, h);
    rmsnorm_kernel<<<4, 256, 0, stream>>>(h, ln2 + (size_t)i * HID, xn);
    // fused Gate/Up: Ntot=11264, K=2048, split-K x4 (slice 512) -> 352 blocks
    gemv_async_kernel<<<dim3(88, 4), 256, 0, stream>>>(
        Wg + i * sWgu, Wu + i * sWgu, Wu + i * sWgu, INTER, INTER, xn, HID, 512, pbuf,
        2 * INTER);
    gu_reduce_kernel<<<88, 256, 0, stream>>>(pbuf, actm);
    // Down: Ntot=2048, K=5632, split-K x11 (slice 512) -> 176 blocks
    gemv_async_kernel<<<dim3(16, 11), 256, 0, stream>>>(
        Wd + i * sWd, Wd + i * sWd, Wd + i * sWd, HID, 0, actm, INTER, 512, pbuf, HID);
    resid_reduce_kernel<<<32, 256, 0, stream>>>(pbuf, 11, h);
  }
}